// GIN_13606456394537
// MI455X (gfx1250) — compile-verified
//
#include <hip/hip_runtime.h>
#include <hip/hip_bf16.h>

typedef __attribute__((ext_vector_type(2))) float v2f;
typedef __attribute__((ext_vector_type(8))) float v8f;

#define D 64
#define XS_STRIDE 68   // 64+4 pad: A b64 read banks (4m+ka..+1) conflict-free, 8B aligned
#define W2_STRIDE 80   // float2 units; 2*80 % 64 == 32 -> half-waves on disjoint bank halves

// ---------------------------------------------------------------------------
// zero-fill
// ---------------------------------------------------------------------------
__global__ void fill_zero_kernel(float* __restrict__ p, int n) {
    int i = blockIdx.x * blockDim.x + threadIdx.x;
    if (i < n) p[i] = 0.0f;
}

// ---------------------------------------------------------------------------
// edge scatter: agg[dst] += h[src]; 16 lanes per edge, float4 (4 ch) per lane
// ---------------------------------------------------------------------------
__global__ void scatter_kernel(const float* __restrict__ H,
                               const long long* __restrict__ ei,
                               float* __restrict__ agg, int nedges) {
    int tid = blockIdx.x * blockDim.x + threadIdx.x;
    int total = nedges * 16;
    if (tid >= total) return;
    int e   = tid >> 4;
    int sub = (tid & 15) << 2;
    int s = (int)ei[e];
    int d = (int)ei[nedges + e];
    float4 v = *(const float4*)(H + (size_t)s * D + sub);
    float* dst = agg + (size_t)d * D + sub;
    unsafeAtomicAdd(dst + 0, v.x);
    unsafeAtomicAdd(dst + 1, v.y);
    unsafeAtomicAdd(dst + 2, v.z);
    unsafeAtomicAdd(dst + 3, v.w);
}

// ---------------------------------------------------------------------------
// WMMA GEMM: Y[nrows x NCOLS] = act( A[nrows x 64] @ W[64 x NCOLS] + bias )
//   A = (1+eps)*X + agg   when FUSE_AGG
//   256 threads / block, 64 rows / block, 8 waves.
//   W is staged K-PAIRED in LDS: Ws2[p][n] = (W[2p][n], W[2p+1][n]) so each
//   B fragment is one aligned ds_load_b64 directly into an even VGPR pair
//   (no packing movs before v_wmma).
//   NCOLS==64: wave -> row-tile (wave&3), col-half (wave>>2)*32, 2 tiles/wave.
//   NCOLS==32: wave -> row-tile (wave&3), col-tile (wave>>2)*16, 1 tile/wave.
//   All control flow in the WMMA region is compile-time or scalar, so EXEC
//   stays all-ones as the WMMA ISA requires.
// ---------------------------------------------------------------------------
template <int NCOLS, bool FUSE_AGG, bool RELU>
__global__ __launch_bounds__(256) void gemm_wmma_kernel(
    const float* __restrict__ X, const float* __restrict__ W,
    const float* __restrict__ bias, float* __restrict__ Y,
    const float* __restrict__ agg, const float* __restrict__ epss, int layer,
    int nrows)
{
    static_assert(NCOLS == 64 || NCOLS == 32, "NCOLS must be 64 or 32");
    constexpr int TILES = (NCOLS == 64) ? 2 : 1;

    __shared__ float Xs[D * XS_STRIDE];
    __shared__ v2f   Ws2[(D / 2) * W2_STRIDE];   // k-paired weights

    const int tid  = threadIdx.x;
    const int row0 = blockIdx.x * 64;

    float eps1 = 1.0f;
    if (FUSE_AGG) eps1 = 1.0f + epss[layer];

    // ---- stage X tile (64 rows x 64 cols), fusing (1+eps)*x + agg ----
    const bool full = (row0 + 64) <= nrows;   // scalar per block
    if (full) {
#pragma unroll
        for (int i = 0; i < 4; ++i) {
            int f  = tid + i * 256;
            int r  = f >> 4;
            int c4 = (f & 15) << 2;
            int gr = row0 + r;
            float4 v = *(const float4*)(X + (size_t)gr * D + c4);
            if (FUSE_AGG) {
                float4 a = *(const float4*)(agg + (size_t)gr * D + c4);
                v.x = eps1 * v.x + a.x;  v.y = eps1 * v.y + a.y;
                v.z = eps1 * v.z + a.z;  v.w = eps1 * v.w + a.w;
            }
            *(float4*)(&Xs[r * XS_STRIDE + c4]) = v;
        }
    } else {
#pragma unroll
        for (int i = 0; i < 4; ++i) {
            int f  = tid + i * 256;
            int r  = f >> 4;
            int c4 = (f & 15) << 2;
            int gr = row0 + r;
            float4 v = make_float4(0.f, 0.f, 0.f, 0.f);
            if (gr < nrows) {
                v = *(const float4*)(X + (size_t)gr * D + c4);
                if (FUSE_AGG) {
                    float4 a = *(const float4*)(agg + (size_t)gr * D + c4);
                    v.x = eps1 * v.x + a.x;  v.y = eps1 * v.y + a.y;
                    v.z = eps1 * v.z + a.z;  v.w = eps1 * v.w + a.w;
                }
            }
            *(float4*)(&Xs[r * XS_STRIDE + c4]) = v;
        }
    }

    // ---- stage W k-paired: Ws2[p][c] = (W[2p][c], W[2p+1][c]) ----
    constexpr int C4PR   = NCOLS / 4;          // float4 groups per row: 16 or 8
    constexpr int GROUPS = (D / 2) * C4PR;     // 512 or 256
#pragma unroll
    for (int i = 0; i < GROUPS / 256; ++i) {
        int g  = tid + i * 256;
        int p  = g / C4PR;
        int c4 = (g % C4PR) << 2;
        float4 v0 = *(const float4*)(W + (size_t)(2 * p)     * NCOLS + c4);
        float4 v1 = *(const float4*)(W + (size_t)(2 * p + 1) * NCOLS + c4);
        v2f* dst = &Ws2[p * W2_STRIDE + c4];
        dst[0] = (v2f){v0.x, v1.x};
        dst[1] = (v2f){v0.y, v1.y};
        dst[2] = (v2f){v0.z, v1.z};
        dst[3] = (v2f){v0.w, v1.w};
    }
    __syncthreads();

    // scalarized wave id -> uniform tile assignment
    const int wave = __builtin_amdgcn_readfirstlane(tid >> 5);
    const int lane = tid & 31;
    const int lo   = lane & 15;
    const int hi   = lane >> 4;

    const int rt = wave & 3;                               // row tile 0..3
    const int n0 = (NCOLS == 64) ? (wave >> 2) * 32        // col base (scalar)
                                 : (wave >> 2) * 16;
    const int m  = rt * 16 + lo;

    const int a_off = m * XS_STRIDE;
    const int b_off = n0 + lo;

    v8f acc0 = {};
    v8f acc1 = {};

#pragma unroll
    for (int k0 = 0; k0 < D; k0 += 4) {
        const int ka = k0 + 2 * hi;            // per half-wave K pair
        const int pr = (k0 >> 1) + hi;         // pair-row in Ws2

        v2f a  = *(const v2f*)(&Xs[a_off + ka]);          // ds_load_b64
        v2f b0 = Ws2[pr * W2_STRIDE + b_off];             // ds_load_b64
        acc0 = __builtin_amdgcn_wmma_f32_16x16x4_f32(
            false, a, false, b0, (short)0, acc0, false, false);

        if (TILES == 2) {
            v2f b1 = Ws2[pr * W2_STRIDE + b_off + 16];    // merges to 2addr_b64
            acc1 = __builtin_amdgcn_wmma_f32_16x16x4_f32(
                false, a, false, b1, (short)0, acc1, false, false);
        }
    }

    // ---- epilogue: bias + relu + store (C/D layout: VGPR r -> M=r+8*hi, N=lo)
    const int gm_base = row0 + rt * 16;
    const int nn = n0 + lo;
    const float bv0 = bias[nn];
    const float bv1 = (TILES == 2) ? bias[nn + 16] : 0.0f;

    if (full) {
#pragma unroll
        for (int r = 0; r < 8; ++r) {
            int gr = gm_base + r + 8 * hi;
            float v = acc0[r] + bv0;
            if (RELU) v = fmaxf(v, 0.0f);
            Y[(size_t)gr * NCOLS + nn] = v;
            if (TILES == 2) {
                float w = acc1[r] + bv1;
                if (RELU) w = fmaxf(w, 0.0f);
                Y[(size_t)gr * NCOLS + nn + 16] = w;
            }
        }
    } else {
#pragma unroll
        for (int r = 0; r < 8; ++r) {
            int gr = gm_base + r + 8 * hi;
            if (gr < nrows) {
                float v = acc0[r] + bv0;
                if (RELU) v = fmaxf(v, 0.0f);
                Y[(size_t)gr * NCOLS + nn] = v;
                if (TILES == 2) {
                    float w = acc1[r] + bv1;
                    if (RELU) w = fmaxf(w, 0.0f);
                    Y[(size_t)gr * NCOLS + nn + 16] = w;
                }
            }
        }
    }
}

// ---------------------------------------------------------------------------
// BN stats: stats[0..63] = sum per channel, stats[64..127] = sum of squares
// ---------------------------------------------------------------------------
__global__ __launch_bounds__(256) void bn_stats_kernel(const float* __restrict__ Z,
                                                       float* __restrict__ stats,
                                                       int nrows) {
    __shared__ float ls[256];
    __shared__ float ls2[256];
    const int t = threadIdx.x;
    const int c = t & 63;
    const int rg = t >> 6;            // 4 rows per block-iteration
    float s = 0.f, s2 = 0.f;
    for (int row = blockIdx.x * 4 + rg; row < nrows; row += gridDim.x * 4) {
        float v = Z[(size_t)row * D + c];
        s += v;  s2 += v * v;
    }
    ls[t] = s;  ls2[t] = s2;
    __syncthreads();
    if (t < 64) {
        s  = ls[t]  + ls[t + 64]  + ls[t + 128]  + ls[t + 192];
        s2 = ls2[t] + ls2[t + 64] + ls2[t + 128] + ls2[t + 192];
        unsafeAtomicAdd(&stats[t], s);
        unsafeAtomicAdd(&stats[64 + t], s2);
    }
}

// ---------------------------------------------------------------------------
// BN apply + relu: h = relu(gamma * (z - mu) * rsqrt(var + eps) + beta)
// ---------------------------------------------------------------------------
__global__ void bn_apply_kernel(const float* __restrict__ Z, float* __restrict__ H,
                                const float* __restrict__ stats,
                                const float* __restrict__ gamma,
                                const float* __restrict__ beta, int nrows) {
    int idx = blockIdx.x * blockDim.x + threadIdx.x;
    int total = nrows * D;
    if (idx >= total) return;
    int c = idx & 63;
    float invN = 1.0f / (float)nrows;
    float mu  = stats[c] * invN;
    float var = stats[64 + c] * invN - mu * mu;
    float sc  = gamma[c] * rsqrtf(var + 1e-5f);
    float v   = (Z[idx] - mu) * sc + beta[c];
    H[idx] = fmaxf(v, 0.0f);
}

// ---------------------------------------------------------------------------
extern "C" void kernel_launch(void* const* d_in, const int* in_sizes, int n_in,
                              void* d_out, int out_size, void* d_ws, size_t ws_size,
                              hipStream_t stream) {
    const float*     x      = (const float*)d_in[0];
    const long long* ei     = (const long long*)d_in[1];   // int64 (2, E)
    const float*     W1s    = (const float*)d_in[2];
    const float*     b1s    = (const float*)d_in[3];
    const float*     W2s    = (const float*)d_in[4];
    const float*     b2s    = (const float*)d_in[5];
    const float*     gammas = (const float*)d_in[6];
    const float*     betas  = (const float*)d_in[7];
    const float*     epss   = (const float*)d_in[8];
    const float*     lin_w  = (const float*)d_in[9];
    const float*     lin_b  = (const float*)d_in[10];
    float*           out    = (float*)d_out;

    const int N = in_sizes[0] / D;       // 100000
    const int E = in_sizes[1] / 2;       // 1200000

    const size_t S = (size_t)N * D;      // floats per node matrix
    float* h     = (float*)d_ws;
    float* agg   = h   + S;              // reused as t2
    float* t1    = agg + S;
    float* stats = t1  + S;              // 128 floats

    const int fill_blocks  = (int)((S + 255) / 256);
    const int scat_blocks  = (E * 16 + 255) / 256;
    const int gemm_blocks  = (N + 63) / 64;
    const int elem_blocks  = (int)((S + 255) / 256);

    for (int layer = 0; layer < 3; ++layer) {
        const float* hin = (layer == 0) ? x : h;

        fill_zero_kernel<<<fill_blocks, 256, 0, stream>>>(agg, (int)S);
        fill_zero_kernel<<<1, 128, 0, stream>>>(stats, 128);

        scatter_kernel<<<scat_blocks, 256, 0, stream>>>(hin, ei, agg, E);

        // t1 = relu(((1+eps)*h + agg) @ W1 + b1)
        gemm_wmma_kernel<64, true, true><<<gemm_blocks, 256, 0, stream>>>(
            hin, W1s + layer * D * D, b1s + layer * D, t1, agg, epss, layer, N);

        // t2 (in agg) = relu(t1 @ W2 + b2)
        gemm_wmma_kernel<64, false, true><<<gemm_blocks, 256, 0, stream>>>(
            t1, W2s + layer * D * D, b2s + layer * D, agg, nullptr, epss, layer, N);

        bn_stats_kernel<<<512, 256, 0, stream>>>(agg, stats, N);

        bn_apply_kernel<<<elem_blocks, 256, 0, stream>>>(
            agg, h, stats, gammas + layer * D, betas + layer * D, N);
    }

    // out = h @ lin_w + lin_b   [N x 64] @ [64 x 32], no relu
    gemm_wmma_kernel<32, false, false><<<gemm_blocks, 256, 0, stream>>>(
        h, lin_w, lin_b, out, nullptr, epss, 0, N);
}